// TaskAlignedAssigner_90915867722337
// MI455X (gfx1250) — compile-verified
//
#include <hip/hip_runtime.h>
#include <hip/hip_bf16.h>
#include <math.h>

// ---------------- problem constants (from reference setup) ----------------
#define BS  2
#define NM  10
#define NA  8400
#define NB  36
#define NP  360
#define NCL 80
#define TOPK 13
#define TA  8            // anchors per k_dist block
#define TILE16 (NA/16)   // 525 wmma tiles per (b,m)

// output flat offsets (floats), reference tuple order
#define O_LAB  0                       // (BS,NA)            16800
#define O_BOX  16800                   // (BS,NA,4)          67200
#define O_SCR  84000                   // (BS,NA,80)       1344000
#define O_MPB  1428000                 // (BS,NM,NA)        168000
#define O_TGI  1596000                 // (BS,NA)            16800
#define O_GTD  1612800                 // (BS,NM,NA,36)    6048000
#define O_CEN  7660800                 // (BS,NM,NA)        168000
#define O_FG   7828800                 // (BS,NA)            16800

// workspace flat offsets (floats)
#define W_OV   0
#define W_AL   168000
#define W_SEL  336000
#define W_MP   504000
#define W_PA   672000                  // pos_align per (b,m): 20
#define W_PO   672020                  // pos_ov per (b,m): 20

typedef float v2f __attribute__((ext_vector_type(2)));
typedef float v8f __attribute__((ext_vector_type(8)));

// ---------------- stage A: dist_max per (b,m,anchor,bin) ----------------
// Compute (dist, angle) once per (anchor,point) in LDS, then 36 bin threads
// scan with a branch-free top-4 insertion. Writes UNMASKED dist_max into the
// gt_dist output region (masked later in place by k_gtdist).
__global__ __launch_bounds__(TA*NB) void k_dist(const float* __restrict__ anc,
                                                const float* __restrict__ coor,
                                                float* __restrict__ dist_out)
{
    __shared__ float px[NP], py[NP];
    __shared__ float sd[TA][NP];
    __shared__ float sa[TA][NP];
    const int tid  = threadIdx.x;            // 0..287
    const int tile = blockIdx.x % (NA/TA);
    const int m    = (blockIdx.x / (NA/TA)) % NM;
    const int b    =  blockIdx.x / ((NA/TA)*NM);
    const int a0   = tile * TA;

    const float* c = coor + (b*NM + m) * (NP*2);
    for (int i = tid; i < NP; i += TA*NB) { px[i] = c[2*i]; py[i] = c[2*i+1]; }
    __syncthreads();

    for (int j = tid; j < TA*NP; j += TA*NB) {
        const int ai = j / NP, p = j % NP;
        const float ax = anc[2*(a0+ai)], ay = anc[2*(a0+ai)+1];
        const float dx = px[p] - ax, dy = py[p] - ay;
        sd[ai][p] = sqrtf(dx*dx + dy*dy);
        float ang = atan2f(dy, dx) * 57.29577951308232f;
        if (ang < 0.f) ang += 360.f;
        sa[ai][p] = ang;
    }
    __syncthreads();

    const int ai = tid / NB, j = tid % NB;
    const float th = (float)(10*j);
    float d0=3e38f,d1=3e38f,d2=3e38f,d3=3e38f;
    float r0=0.f,r1=0.f,r2=0.f,r3=0.f;
    for (int p = 0; p < NP; ++p) {
        float diff = fabsf(sa[ai][p] - th);
        diff = diff > 180.f ? 360.f - diff : diff;
        const float r = sd[ai][p];
        const bool c0 = diff < d0, c1 = diff < d1, c2 = diff < d2, c3 = diff < d3;
        const float nd3 = c3 ? (c2 ? d2 : diff) : d3, nr3 = c3 ? (c2 ? r2 : r) : r3;
        const float nd2 = c2 ? (c1 ? d1 : diff) : d2, nr2 = c2 ? (c1 ? r1 : r) : r2;
        const float nd1 = c1 ? (c0 ? d0 : diff) : d1, nr1 = c1 ? (c0 ? r0 : r) : r1;
        const float nd0 = c0 ? diff : d0,             nr0 = c0 ? r : r0;
        d0=nd0; d1=nd1; d2=nd2; d3=nd3; r0=nr0; r1=nr1; r2=nr2; r3=nr3;
    }
    float out = (d0 > 3.0f) ? 1e-6f : fmaxf(fmaxf(r0,r1), fmaxf(r2,r3));
    out = fmaxf(out, 1e-6f);
    dist_out[(((b*NM + m)*NA) + (a0+ai))*NB + j] = out;
}

// ---------------- stage B: mask-IoU via V_WMMA_F32_16X16X4_F32 ----------------
// One wave per 16-anchor tile of one (b,m). Row-sums of min/max over 36 rays
// done on the matrix pipe: A = 16x4 value chunk, B = ones -> D[M,*] = rowsum.
__global__ __launch_bounds__(128) void k_overlap(const float* __restrict__ pred,
                                                 const float* __restrict__ scores,
                                                 const float* __restrict__ anc,
                                                 const int*   __restrict__ gl,
                                                 const float* __restrict__ gtb,
                                                 const float* __restrict__ mgt,
                                                 const float* __restrict__ dist,
                                                 float* __restrict__ ws)
{
    __shared__ float smin[4][16];
    __shared__ float smax[4][16];
    const int tid  = threadIdx.x;
    const int wv   = tid >> 5;
    const int l    = tid & 31;
    const int t    = blockIdx.x * 4 + wv;       // 0..10499 (exact)
    const int a0   = (t % TILE16) * 16;
    const int m    = (t / TILE16) % NM;
    const int b    =  t / (TILE16 * NM);
    const int lr   = l & 15;
    const int kb   = (l >> 4) * 2;              // A layout: low lanes K=c+0..1, high lanes K=c+2..3
    const int a    = a0 + lr;

    const float* trow = dist + ((b*NM + m)*NA + a) * NB;
    const float* prow = pred + (b*NA + a) * NB;
    __builtin_prefetch(prow, 0, 1);             // global_prefetch_b8

    v8f cmin = {}; v8f cmax = {};
    v2f ones; ones.x = 1.0f; ones.y = 1.0f;
#pragma unroll
    for (int c = 0; c < NB; c += 4) {
        const float t0 = trow[c+kb], t1 = trow[c+kb+1];
        const float p0 = prow[c+kb], p1 = prow[c+kb+1];
        v2f amin, amax;
        amin.x = fmaxf(fminf(t0,p0), 1e-6f);
        amin.y = fmaxf(fminf(t1,p1), 1e-6f);
        amax.x = fmaxf(t0,p0);
        amax.y = fmaxf(t1,p1);
        cmin = __builtin_amdgcn_wmma_f32_16x16x4_f32(false, amin, false, ones, (short)0, cmin, false, false);
        cmax = __builtin_amdgcn_wmma_f32_16x16x4_f32(false, amax, false, ones, (short)0, cmax, false, false);
    }
    // D layout: VGPR v @ lane<16 -> M=v ; @ lane>=16 -> M=v+8 (replicated over N)
    if (l == 0) {
#pragma unroll
        for (int v = 0; v < 8; ++v) { smin[wv][v] = cmin[v]; smax[wv][v] = cmax[v]; }
    }
    if (l == 16) {
#pragma unroll
        for (int v = 0; v < 8; ++v) { smin[wv][8+v] = cmin[v]; smax[wv][8+v] = cmax[v]; }
    }
    __syncthreads();

    if (l < 16) {
        const int   idx = (b*NM + m)*NA + a;
        const float S1  = smin[wv][l];
        const float S2  = smax[wv][l];
        const float ovr = S1 / S2;
        const float ax = anc[2*a], ay = anc[2*a+1];
        const float* gb = gtb + (b*NM + m)*4;
        const float dmin = fminf(fminf(ax - gb[0], ay - gb[1]), fminf(gb[2] - ax, gb[3] - ay));
        const bool sel = (dmin > 1e-9f) && (mgt[b*NM + m] > 0.f);
        const float ov  = sel ? ovr : 0.f;
        const int   lab = gl[b*NM + m];
        const float sc  = scores[(b*NA + a)*NCL + lab];
        const float bsc = sel ? sc : 0.f;
        ws[W_OV  + idx] = ov;
        ws[W_AL  + idx] = bsc * ov * ov * ov;   // score^1 * overlap^3
        ws[W_SEL + idx] = sel ? 1.f : 0.f;
        ws[W_MP  + idx] = 0.f;                  // pre-zero mask_pos each call
    }
}

// ---------------- stage C: top-13 per (b,m) over 8400 anchors ----------------
__global__ __launch_bounds__(256) void k_topk(float* __restrict__ ws)
{
    __shared__ float sv[NA];
    __shared__ float rv[256];
    __shared__ int   ri[256];
    const int tid = threadIdx.x;
    const int bm  = blockIdx.x;                 // 0..19
    const float* arow = ws + W_AL + bm*NA;
    for (int i = tid; i < NA; i += 256) sv[i] = arow[i];
    __syncthreads();
    for (int k = 0; k < TOPK; ++k) {
        float bv = -3e38f; int bi = 0x7fffffff;
        for (int i = tid; i < NA; i += 256) {
            const float v = sv[i];
            if (v > bv || (v == bv && i < bi)) { bv = v; bi = i; }
        }
        rv[tid] = bv; ri[tid] = bi;
        __syncthreads();
        for (int s = 128; s > 0; s >>= 1) {
            if (tid < s) {
                const float v2 = rv[tid+s]; const int i2 = ri[tid+s];
                if (v2 > rv[tid] || (v2 == rv[tid] && i2 < ri[tid])) { rv[tid] = v2; ri[tid] = i2; }
            }
            __syncthreads();
        }
        if (tid == 0) {
            const int w = ri[0];
            sv[w] = -3e38f;
            ws[W_MP + bm*NA + w] = ws[W_SEL + bm*NA + w];
        }
        __syncthreads();
    }
}

// ---------------- stage D: multi-GT resolution + per-anchor outputs ----------------
__global__ __launch_bounds__(256) void k_resolve(const int* __restrict__ gl,
                                                 const float* __restrict__ gtb,
                                                 float* __restrict__ ws,
                                                 float* __restrict__ out)
{
    const int i = blockIdx.x * 256 + threadIdx.x;
    if (i >= BS*NA) return;
    const int b = i / NA, a = i % NA;
    float mp[NM], ov[NM];
    float fg = 0.f;
#pragma unroll
    for (int m = 0; m < NM; ++m) {
        const int idx = (b*NM + m)*NA + a;
        mp[m] = ws[W_MP + idx];
        ov[m] = ws[W_OV + idx];
        fg += mp[m];
    }
    if (fg > 1.f) {
        int mb = 0; float bv = ov[0];
#pragma unroll
        for (int m = 1; m < NM; ++m) if (ov[m] > bv) { bv = ov[m]; mb = m; }
#pragma unroll
        for (int m = 0; m < NM; ++m) mp[m] = (m == mb) ? 1.f : 0.f;
    }
    float fgm = 0.f; int tgi = 0; bool found = false;
#pragma unroll
    for (int m = 0; m < NM; ++m) {
        if (!found && mp[m] > 0.f) { tgi = m; found = true; }
        fgm += mp[m];
        const int idx = (b*NM + m)*NA + a;
        ws[W_MP + idx] = mp[m];
        out[O_MPB + idx] = mp[m];
    }
    out[O_FG  + i] = (fgm > 0.f) ? 1.f : 0.f;
    out[O_TGI + i] = (float)tgi;
    int lab = gl[b*NM + tgi]; if (lab < 0) lab = 0;
    out[O_LAB + i] = (float)lab;
#pragma unroll
    for (int k = 0; k < 4; ++k) out[O_BOX + i*4 + k] = gtb[(b*NM + tgi)*4 + k];
}

// ---------------- stage E1: per-(b,m) max reductions ----------------
__global__ __launch_bounds__(256) void k_rowred(float* __restrict__ ws)
{
    __shared__ float ra[256], ro[256];
    const int tid = threadIdx.x;
    const int bm  = blockIdx.x;
    float ba = 0.f, bo = 0.f;
    for (int i = tid; i < NA; i += 256) {
        const int idx = bm*NA + i;
        const float mm = ws[W_MP + idx];
        ba = fmaxf(ba, ws[W_AL + idx] * mm);
        bo = fmaxf(bo, ws[W_OV + idx] * mm);
    }
    ra[tid] = ba; ro[tid] = bo;
    __syncthreads();
    for (int s = 128; s > 0; s >>= 1) {
        if (tid < s) { ra[tid] = fmaxf(ra[tid], ra[tid+s]); ro[tid] = fmaxf(ro[tid], ro[tid+s]); }
        __syncthreads();
    }
    if (tid == 0) { ws[W_PA + bm] = ra[0]; ws[W_PO + bm] = ro[0]; }
}

// ---------------- stage E2: normalized one-hot target scores ----------------
__global__ __launch_bounds__(256) void k_scores(const float* __restrict__ ws,
                                                float* __restrict__ out)
{
    const int i = blockIdx.x * 256 + threadIdx.x;
    if (i >= BS*NA) return;
    const int b = i / NA, a = i % NA;
    float norm = 0.f;
#pragma unroll
    for (int m = 0; m < NM; ++m) {
        const int idx = (b*NM + m)*NA + a;
        const float am = ws[W_AL + idx] * ws[W_MP + idx];
        norm = fmaxf(norm, am * ws[W_PO + b*NM + m] / (ws[W_PA + b*NM + m] + 1e-9f));
    }
    const int   lab = (int)out[O_LAB + i];
    const float val = (out[O_FG + i] > 0.f) ? norm : 0.f;
    float* row = out + O_SCR + (size_t)i * NCL;
    for (int c = 0; c < NCL; ++c) row[c] = (c == lab) ? val : 0.f;
}

// ---------------- stage F: mask gt_dist in place + centerness ----------------
__global__ __launch_bounds__(256) void k_gtdist(const float* __restrict__ ws,
                                                float* __restrict__ out)
{
    const int i = blockIdx.x * 256 + threadIdx.x;
    if (i >= BS*NM*NA) return;
    const float mask = ws[W_MP + i];
    const int base = O_GTD + i*NB;
    float mn = 3e38f, mx = 0.f;
#pragma unroll
    for (int k = 0; k < NB; ++k) {
        const float d = out[base + k];
        mn = fminf(mn, d); mx = fmaxf(mx, d);
        out[base + k] = (mask > 0.f) ? d : 0.f;
    }
    out[O_CEN + i] = (mask > 0.f) ? sqrtf(mn / mx) : 0.f;
}

// ---------------- host-side launcher ----------------
extern "C" void kernel_launch(void* const* d_in, const int* in_sizes, int n_in,
                              void* d_out, int out_size, void* d_ws, size_t ws_size,
                              hipStream_t stream)
{
    const float* pd_scores = (const float*)d_in[0];  // (BS,NA,80)
    const float* pd_bboxes = (const float*)d_in[1];  // (BS,NA,36)
    const float* anc       = (const float*)d_in[2];  // (NA,2)
    const int*   gt_labels = (const int*)  d_in[3];  // (BS,NM,1)
    const float* gt_bboxes = (const float*)d_in[4];  // (BS,NM,4)
    const float* mask_gt   = (const float*)d_in[5];  // (BS,NM,1)
    const float* gt_coor   = (const float*)d_in[6];  // (BS,NM,720)
    float* out = (float*)d_out;
    float* ws  = (float*)d_ws;

    // A: dist_max -> out[O_GTD] (unmasked for now)
    k_dist<<<BS*NM*(NA/TA), TA*NB, 0, stream>>>(anc, gt_coor, out + O_GTD);
    // B: WMMA mask-IoU reductions -> overlaps/align/sel, zero mask_pos
    k_overlap<<<(BS*NM*TILE16)/4, 128, 0, stream>>>(pd_bboxes, pd_scores, anc,
                                                    gt_labels, gt_bboxes, mask_gt,
                                                    out + O_GTD, ws);
    // C: top-13 per (b,m)
    k_topk<<<BS*NM, 256, 0, stream>>>(ws);
    // D: multi-GT resolution + labels/boxes/fg/idx/mask outputs
    k_resolve<<<(BS*NA + 255)/256, 256, 0, stream>>>(gt_labels, gt_bboxes, ws, out);
    // E1: per-(b,m) pos_align / pos_ov
    k_rowred<<<BS*NM, 256, 0, stream>>>(ws);
    // E2: target_scores
    k_scores<<<(BS*NA + 255)/256, 256, 0, stream>>>(ws, out);
    // F: mask gt_dist in place + centerness
    k_gtdist<<<(BS*NM*NA + 255)/256, 256, 0, stream>>>(ws, out);
}